// EdgeLoss_6786048328604
// MI455X (gfx1250) — compile-verified
//
#include <hip/hip_runtime.h>
#include <math.h>
#include <stdint.h>

// ---------------------------------------------------------------------------
// Fused Canny edge-loss for MI455X (gfx1250).
//   targets/outputs: [4,16,3,256,256] f32  ->  scalar mean |edge(o)-edge(t)|
//
// One workgroup = one 32x32 output tile of one frame. Whole pipeline
// (gauss blur -> sobel -> mag/orientation -> NMS -> threshold) fused in LDS.
// Input tiles are pulled global->LDS with the CDNA5 Tensor Data Mover
// (tensor_load_to_lds, TENSORcnt), one 2D DMA per channel per pass.
// ---------------------------------------------------------------------------

typedef unsigned int u32x4 __attribute__((ext_vector_type(4)));
typedef int          i32x8 __attribute__((ext_vector_type(8)));
typedef int          i32x4 __attribute__((ext_vector_type(4)));

#define IMG_H 256
#define IMG_W 256
#define TILE  32
#define HALO  4              // 2 (blur) + 1 (sobel) + 1 (NMS)
#define INW   (TILE + 2*HALO)   // 40
#define HBW   36                // hblur cols: [-2..33]
#define BLW   36                // blurred:    [-2..33]^2
#define MGW   34                // mag/gx/gy:  [-1..32]^2
#define NFRAMES 64
#define TPB   256

__device__ __forceinline__ void tdm_load_tile_2d(const float* gsrc,
                                                 float* s_in,   // LDS dst
                                                 int org_r, int org_c,
                                                 int lh, int lw) {
    // Build a 2-group Tensor DMA descriptor (D#) for an lh x lw f32 tile of a
    // 256x256 image, row stride IMG_W, packed into LDS at &s_in[0].
    unsigned long long gaddr =
        (unsigned long long)(uintptr_t)(gsrc + (size_t)org_r * IMG_W + org_c);
    // flat "shared" pointer: low 32 bits are the LDS byte address
    unsigned lds_addr = (unsigned)(uintptr_t)s_in;

    u32x4 g0;
    g0.x = 1u;                                         // count=1, user mode
    g0.y = lds_addr;                                   // lds_addr[31:0]
    g0.z = (unsigned)(gaddr & 0xffffffffull);          // global_addr[31:0]
    g0.w = (unsigned)((gaddr >> 32) & 0x01ffffffull)   // global_addr[56:32]
         | (2u << 30);                                 // type=2 ("image")

    unsigned td0 = (unsigned)(IMG_W - org_c);          // remaining cols (OOB)
    unsigned td1 = (unsigned)(IMG_H - org_r);          // remaining rows (OOB)
    i32x8 g1;
    g1[0] = (int)(2u << 16);                           // data_size=4B, no flags
    g1[1] = (int)((td0 & 0xffffu) << 16);              // tensor_dim0[15:0]
    g1[2] = (int)(((td0 >> 16) & 0xffffu) | ((td1 & 0xffffu) << 16));
    g1[3] = (int)(((td1 >> 16) & 0xffffu) | ((unsigned)lw << 16)); // tile_dim0
    g1[4] = (int)(unsigned)lh;                         // tile_dim1, tile_dim2=0
    g1[5] = (int)(unsigned)IMG_W;                      // tensor_dim0_stride lo
    g1[6] = 0;
    g1[7] = 0;

    i32x4 zg4 = {0, 0, 0, 0};                          // groups 2/3 unused (2D)
    i32x8 zg8 = {0, 0, 0, 0, 0, 0, 0, 0};
    __builtin_amdgcn_tensor_load_to_lds(g0, g1, zg4, zg4, zg8, 0);
    __builtin_amdgcn_s_wait_tensorcnt(0);
}

// One full Canny pass for this tile of one tensor; fills thr[4] with the
// thresholded thin-edge value of this thread's 4 output pixels.
__device__ __forceinline__ void canny_pass(const float* img,   // [64,3,256,256]
                                           int frame, int tile_r, int tile_c,
                                           float* s_in, float* s_hb, float* s_bl,
                                           float* s_mag, float* s_gx, float* s_gy,
                                           float thr[4]) {
    const int tid = threadIdx.x;
    int org_r = tile_r - HALO; if (org_r < 0) org_r = 0;
    int org_c = tile_c - HALO; if (org_c < 0) org_c = 0;
    const int lh = min(INW, IMG_H - org_r);
    const int lw = min(INW, IMG_W - org_c);

    const float gw0 = 0.13533528323661270f;   // exp(-2)
    const float gw1 = 0.60653065971263342f;   // exp(-0.5)
    const float gw[5] = {gw0, gw1, 1.0f, gw1, gw0};

    __syncthreads();                        // previous pass done reading LDS
    for (int i = tid; i < MGW * MGW; i += TPB) {
        s_mag[i] = 0.0f; s_gx[i] = 0.0f; s_gy[i] = 0.0f;
    }

    for (int ch = 0; ch < 3; ++ch) {
        __syncthreads();                    // s_in consumers of prev ch done
        if (tid < 32) {                     // one wave issues the DMA
            const float* src = img + (size_t)(frame * 3 + ch) * IMG_H * IMG_W;
            tdm_load_tile_2d(src, s_in, org_r, org_c, lh, lw);
        }
        __syncthreads();                    // tile visible to all waves

        // horizontal gaussian (pad (0,2)), rows [-4..35] x cols [-2..33]
        for (int i = tid; i < INW * HBW; i += TPB) {
            int hr = i / HBW, hc = i - hr * HBW;
            int r = tile_r + hr - 4, c = tile_c + hc - 2;
            float acc = 0.0f;
            if ((unsigned)r < IMG_H) {
                const int rr = (r - org_r) * lw - org_c;
#pragma unroll
                for (int k = 0; k < 5; ++k) {
                    int cc = c + k - 2;
                    if ((unsigned)cc < IMG_W) acc += gw[k] * s_in[rr + cc];
                }
            }
            s_hb[i] = acc;
        }
        __syncthreads();

        // vertical gaussian (pad (2,0)); forced 0 outside image so the next
        // conv's zero-padding of "blurred" matches the reference exactly.
        for (int i = tid; i < BLW * BLW; i += TPB) {
            int br = i / BLW, bc = i - br * BLW;   // local (br-2, bc-2)
            int r = tile_r + br - 2, c = tile_c + bc - 2;
            float acc = 0.0f;
            if ((unsigned)r < IMG_H && (unsigned)c < IMG_W) {
#pragma unroll
                for (int k = 0; k < 5; ++k) acc += gw[k] * s_hb[(br + k) * HBW + bc];
            }
            s_bl[i] = acc;
        }
        __syncthreads();

        // sobel + accumulate magnitude / summed gradients (34x34, halo 1).
        // Element i is owned by the same thread every channel -> no races.
        for (int i = tid; i < MGW * MGW; i += TPB) {
            int mr = i / MGW, mc = i - mr * MGW;   // local (mr-1, mc-1)
            int r = tile_r + mr - 1, c = tile_c + mc - 1;
            if ((unsigned)r < IMG_H && (unsigned)c < IMG_W) {
                const float* B = s_bl + mr * BLW + mc;   // local (mr-2.. , ..)
                float a00 = B[0],       a01 = B[1],        a02 = B[2];
                float a10 = B[BLW],                        a12 = B[BLW + 2];
                float a20 = B[2 * BLW], a21 = B[2*BLW + 1], a22 = B[2*BLW + 2];
                float gx = (a00 - a02) + 2.0f * (a10 - a12) + (a20 - a22);
                float gy = (a00 + 2.0f * a01 + a02) - (a20 + 2.0f * a21 + a22);
                s_mag[i] += sqrtf(gx * gx + gy * gy);
                s_gx[i]  += gx;
                s_gy[i]  += gy;
            }
        }
    }
    __syncthreads();

    // NMS + threshold, 4 output pixels per thread
    const int drt[8] = {0, 1, 1, 1, 0, -1, -1, -1};
    const int dct[8] = {1, 1, 0, -1, -1, -1, 0, 1};
#pragma unroll
    for (int k = 0; k < 4; ++k) {
        int p = tid + k * TPB;
        int orow = p >> 5, ocol = p & 31;
        int mi = (orow + 1) * MGW + (ocol + 1);
        float m  = s_mag[mi];
        float gx = s_gx[mi], gy = s_gy[mi];
        float ori = atan2f(gy, gx) * (180.0f / 3.14159265358979323846f) + 180.0f;
        int q  = (int)rintf(ori * (1.0f / 45.0f));     // round-half-even
        int ip = q & 7, in_ = (q + 4) & 7;             // q in [0,8]
        float pos = m - s_mag[(orow + 1 + drt[ip]) * MGW + (ocol + 1 + dct[ip])];
        float neg = m - s_mag[(orow + 1 + drt[in_]) * MGW + (ocol + 1 + dct[in_])];
        float thin = (fminf(pos, neg) > 0.0f) ? m : 0.0f;
        thr[k] = (thin < 2.0f) ? 0.0f : thin;
    }
}

__global__ __launch_bounds__(TPB)
void canny_diff_kernel(const float* __restrict__ targets,
                       const float* __restrict__ outputs,
                       float* __restrict__ partials) {
    __shared__ float s_in [INW * INW];   // 40x40 input tile
    __shared__ float s_hb [INW * HBW];   // 40x36 hblur
    __shared__ float s_bl [BLW * BLW];   // 36x36 blurred
    __shared__ float s_mag[MGW * MGW];   // 34x34 grad magnitude (summed ch)
    __shared__ float s_gx [MGW * MGW];
    __shared__ float s_gy [MGW * MGW];

    const int tid    = threadIdx.x;
    const int frame  = blockIdx.x >> 6;        // 64 tiles per frame
    const int tileId = blockIdx.x & 63;
    const int tile_r = (tileId >> 3) * TILE;
    const int tile_c = (tileId & 7)  * TILE;

    float t0[4], t1[4];
    canny_pass(targets, frame, tile_r, tile_c,
               s_in, s_hb, s_bl, s_mag, s_gx, s_gy, t0);
    canny_pass(outputs, frame, tile_r, tile_c,
               s_in, s_hb, s_bl, s_mag, s_gx, s_gy, t1);

    float acc = 0.0f;
#pragma unroll
    for (int k = 0; k < 4; ++k) acc += fabsf(t1[k] - t0[k]);

    __syncthreads();                // NMS readers done -> reuse s_in as scratch
    s_in[tid] = acc;
    __syncthreads();
#pragma unroll
    for (int s = TPB / 2; s > 0; s >>= 1) {
        if (tid < s) s_in[tid] += s_in[tid + s];
        __syncthreads();
    }
    if (tid == 0) partials[blockIdx.x] = s_in[0];
}

__global__ __launch_bounds__(TPB)
void reduce_mean_kernel(const float* __restrict__ partials,
                        float* __restrict__ out, int n, float inv_count) {
    __shared__ float sm[TPB];
    float a = 0.0f;
    for (int i = threadIdx.x; i < n; i += TPB) a += partials[i];
    sm[threadIdx.x] = a;
    __syncthreads();
#pragma unroll
    for (int s = TPB / 2; s > 0; s >>= 1) {
        if (threadIdx.x < s) sm[threadIdx.x] += sm[threadIdx.x + s];
        __syncthreads();
    }
    if (threadIdx.x == 0) out[0] = sm[0] * inv_count;
}

extern "C" void kernel_launch(void* const* d_in, const int* in_sizes, int n_in,
                              void* d_out, int out_size, void* d_ws, size_t ws_size,
                              hipStream_t stream) {
    const float* targets = (const float*)d_in[0];
    const float* outputs = (const float*)d_in[1];
    float* partials = (float*)d_ws;            // 4096 floats
    float* out      = (float*)d_out;

    const int nblocks = NFRAMES * 64;          // 4096
    canny_diff_kernel<<<nblocks, TPB, 0, stream>>>(targets, outputs, partials);
    reduce_mean_kernel<<<1, TPB, 0, stream>>>(
        partials, out, nblocks,
        1.0f / (float)(NFRAMES * IMG_H * IMG_W));
    (void)in_sizes; (void)n_in; (void)out_size; (void)ws_size;
}